// GroupConv1d_37744172597291
// MI455X (gfx1250) — compile-verified
//
#include <hip/hip_runtime.h>

// TSM temporal 3-tap conv: out[n,t,c,s] = w0*x[n,t-1,c,s] + w1*x[n,t,c,s] + w2*x[n,t+1,c,s]
// Memory-bound (0.62 FLOP/B, ~411 MB traffic -> ~18us @ 23.3 TB/s HBM).
// Cast as D = T2 * X2 with T2 = diag(T(c), T(c)) (16x16, two batches packed
// block-diagonally) so a chain of 4x V_WMMA_F32_16X16X4_F32 (K=16) produces a
// fully dense 16(t,n) x 16(spatial) output tile with zero extra memory traffic.
//
// Grid is 3D (spatial-slab, channel, batch-pair) so there is no integer
// div/mod in the kernel; each wave loops over 7 consecutive spatial tiles,
// reusing its A fragments and weights and keeping several independent
// load->wmma->store chains in flight.

typedef __attribute__((ext_vector_type(2))) float v2f;
typedef __attribute__((ext_vector_type(8))) float v8f;

namespace {
constexpr int C_    = 256;
constexpr int HW_   = 56 * 56;          // 3136
constexpr int CHW_  = C_ * HW_;         // stride between consecutive nt
constexpr int SBLK  = 16;               // spatial columns per tile
constexpr int NSB   = HW_ / SBLK;       // 196 spatial tiles (exact)
constexpr int NPAIR = 4;                // 8 batches packed in pairs
constexpr int WPB   = 4;                // waves per block (blockDim = 128)
constexpr int GX    = 7;                // grid.x
constexpr int ITER  = NSB / (GX * WPB); // 7 tiles per wave: 7*4*7 = 196 exact
}

__device__ __forceinline__ float t2val(int M, int K, float w0, float w1, float w2) {
  // block-diagonal tridiagonal: T2[M,K] = T[M&7, K&7] iff same 8-block
  const int t = M & 7, k = K & 7;
  float r = (k == t + 1) ? w2 : 0.0f;
  r       = (k == t)     ? w1 : r;
  r       = (k == t - 1) ? w0 : r;
  return ((M >> 3) == (K >> 3)) ? r : 0.0f;
}

__global__ __launch_bounds__(128)
void tsm_temporal_conv_wmma(const float* __restrict__ x,
                            const float* __restrict__ w,
                            float* __restrict__ out)
{
  const int lane = threadIdx.x & 31;
  const int wid  = threadIdx.x >> 5;
  const int c    = blockIdx.y;               // channel (block-uniform)
  const int p    = blockIdx.z;               // batch pair -> nt = p*16 + K

  const int half  = lane >> 4;               // 0: lanes 0-15, 1: lanes 16-31
  const int sb0   = (blockIdx.x * WPB + wid) * ITER;
  int colbase     = c * HW_ + sb0 * SBLK + (lane & 15);

  const float w0 = w[3 * c + 0];
  const float w1 = w[3 * c + 1];
  const float w2 = w[3 * c + 2];

#if defined(__AMDGCN__) && __has_builtin(__builtin_amdgcn_wmma_f32_16x16x4_f32)
  // ---- CDNA5 WMMA path ----
  const int M = lane & 15;                   // A-fragment row for this lane

  // Build the 4 A fragments once; reused across all ITER spatial tiles.
  v2f a[4];
  #pragma unroll
  for (int kg = 0; kg < 4; ++kg) {
    const int k0 = kg * 4 + half * 2;        // A 16x4: VGPR0=K0/K2, VGPR1=K1/K3
    a[kg].x = t2val(M, k0 + 0, w0, w1, w2);
    a[kg].y = t2val(M, k0 + 1, w0, w1, w2);
  }

  #pragma unroll
  for (int it = 0; it < ITER; ++it) {
    v8f acc = {0.f, 0.f, 0.f, 0.f, 0.f, 0.f, 0.f, 0.f};
    #pragma unroll
    for (int kg = 0; kg < 4; ++kg) {
      const int k0 = kg * 4 + half * 2;
      v2f b;                                 // B 4x16: same K-half split, N = lane&15
      b.x = __builtin_nontemporal_load(x + (p * 16 + k0 + 0) * CHW_ + colbase);
      b.y = __builtin_nontemporal_load(x + (p * 16 + k0 + 1) * CHW_ + colbase);
      acc = __builtin_amdgcn_wmma_f32_16x16x4_f32(
          /*neg_a=*/false, a[kg], /*neg_b=*/false, b,
          /*c_mod=*/(short)0, acc, /*reuse_a=*/false, /*reuse_b=*/false);
    }
    // D 16x16: VGPR r -> row M = r + 8*half, col = lane&15 => nt = p*16 + M
    #pragma unroll
    for (int r = 0; r < 8; ++r) {
      const int nt = p * 16 + half * 8 + r;
      __builtin_nontemporal_store(acc[r], out + nt * CHW_ + colbase);
    }
    colbase += SBLK;
  }
#else
  // ---- fallback: same tile ownership, streaming 3-tap in registers ----
  const int ntbase = p * 16 + half * 8;      // this lane owns one temporal column
  for (int it = 0; it < ITER; ++it) {
    float prev = 0.0f;
    float cur  = x[ntbase * CHW_ + colbase];
    #pragma unroll
    for (int t = 0; t < 8; ++t) {
      const float nxt = (t < 7) ? x[(ntbase + t + 1) * CHW_ + colbase] : 0.0f;
      out[(ntbase + t) * CHW_ + colbase] = w0 * prev + w1 * cur + w2 * nxt;
      prev = cur;
      cur  = nxt;
    }
    colbase += SBLK;
  }
#endif
}

extern "C" void kernel_launch(void* const* d_in, const int* in_sizes, int n_in,
                              void* d_out, int out_size, void* d_ws, size_t ws_size,
                              hipStream_t stream) {
  (void)in_sizes; (void)n_in; (void)out_size; (void)d_ws; (void)ws_size;
  const float* x = (const float*)d_in[0];
  const float* w = (const float*)d_in[1];
  float* out     = (float*)d_out;

  // 3D grid: no div/mod in kernel. 7*4 waves * 7 tiles = 196 spatial tiles.
  dim3 grid(GX, C_, NPAIR);
  tsm_temporal_conv_wmma<<<grid, WPB * 32, 0, stream>>>(x, w, out);
}